// GuidedFilter_70720931496762
// MI455X (gfx1250) — compile-verified
//
#include <hip/hip_runtime.h>
#include <cmath>

#define GF_H 1024
#define GF_W 1024
#define GF_R 20
#define GF_PL (GF_H * GF_W)   // plane size in elements (fits int32 for <=18 planes)

typedef __attribute__((ext_vector_type(2))) float v2f;
typedef __attribute__((ext_vector_type(8))) float v8f;

__device__ __forceinline__ float gf_inv_cnt(int i, int L) {
    int a = i < GF_R ? i : GF_R;
    int b = (L - 1 - i) < GF_R ? (L - 1 - i) : GF_R;
    return 1.0f / (float)(a + b + 1);
}

__device__ __forceinline__ float gf_softplus(float x) {
    return fmaxf(x, 0.0f) + log1pf(expf(-fabsf(x)));
}

// ---------------------------------------------------------------------------
// Vertical 41-tap box SUM of NCH planes over one 16x16 tile, computed as a
// banded-ones matmul with V_WMMA_F32_16X16X4_F32:
//   D[m][n] = sum_g band(obase+m, g) * in[g][cb+n]
// K tiled over row-blocks kb in [ty-2, ty+2], 4 k-steps of 4. Edge truncation
// = clamped kb range + zeroed band entries.
// f32 layouts (ISA 7.12.2): A 16x4: m=lane&15, k={0,1}/{2,3} by lane half.
// B 4x16: n=lane&15, same k-halves. D: VGPR v -> row v / v+8, col=lane&15.
// All loads use uniform plane base + 32-bit lane offset (SADDR form).
// ---------------------------------------------------------------------------
template <int NCH>
__device__ __forceinline__ void gf_vert_box_tile(const float* __restrict__ in,
                                                 int obase, int cb, int lane,
                                                 v8f (&acc)[NCH]) {
    const int nlo = lane & 15;
    const int khalf = lane >> 4;
    const int colo = cb + nlo;
#pragma unroll
    for (int ch = 0; ch < NCH; ++ch)
        acc[ch] = v8f{0.f, 0.f, 0.f, 0.f, 0.f, 0.f, 0.f, 0.f};

#if defined(__HIP_DEVICE_COMPILE__) && __has_builtin(__builtin_amdgcn_wmma_f32_16x16x4_f32)
    const int ty = obase >> 4;
    const int kbmax = (GF_H / 16) - 1;
    int kb0 = ty - 2; if (kb0 < 0) kb0 = 0;
    int kb1 = ty + 2; if (kb1 > kbmax) kb1 = kbmax;
    for (int kb = kb0; kb <= kb1; ++kb) {       // wave-uniform bounds
#pragma unroll
        for (int q = 0; q < 4; ++q) {
            const int r0 = kb * 16 + q * 4 + 2 * khalf;  // input row of k-elem 0
            const int d0 = (obase + nlo) - r0;           // band distance, A[m][k]
            const int d1 = d0 - 1;
            v2f a;
            a.x = (d0 <= GF_R && d0 >= -GF_R) ? 1.0f : 0.0f;
            a.y = (d1 <= GF_R && d1 >= -GF_R) ? 1.0f : 0.0f;
            const int off = r0 * GF_W + colo;            // lane-varying i32
#pragma unroll
            for (int ch = 0; ch < NCH; ++ch) {
                const float* pch = in + ch * GF_PL;      // wave-uniform base
                v2f b;
                b.x = pch[off];
                b.y = pch[off + GF_W];
                acc[ch] = __builtin_amdgcn_wmma_f32_16x16x4_f32(
                    false, a, false, b, (short)0, acc[ch], false, false);
            }
        }
    }
#else
    // Scalar fallback (host pass / missing builtin): identical lane mapping.
#pragma unroll
    for (int ch = 0; ch < NCH; ++ch) {
#pragma unroll
        for (int v = 0; v < 8; ++v) {
            const int row = obase + v + 8 * khalf;
            int g0 = row - GF_R; if (g0 < 0) g0 = 0;
            int g1 = row + GF_R; if (g1 > GF_H - 1) g1 = GF_H - 1;
            float s = 0.f;
            for (int g = g0; g <= g1; ++g)
                s += in[ch * GF_PL + g * GF_W + colo];
            acc[ch][v] = s;
        }
    }
#endif
}

// ---------------------------------------------------------------------------
// K1: horizontal box MEAN of [x(3), y(3), x*x(3), y(i)*x(j)(9)] for one row.
// Products formed on the fly from LDS-staged rows; sliding window per thread
// over 4 contiguous columns; float4 coalesced stores.
// ---------------------------------------------------------------------------
__global__ void __launch_bounds__(256)
gf_k1_hmean18(const float* __restrict__ xn, const float* __restrict__ yn,
              float* __restrict__ o18) {
    __shared__ float rows[6][GF_W];
    const int row = blockIdx.x;
    const int t = threadIdx.x;
#pragma unroll
    for (int c = 0; c < 3; ++c) {
        const float4* px = (const float4*)(xn + c * GF_PL + row * GF_W);
        ((float4*)rows[c])[t] = px[t];
        const float4* py = (const float4*)(yn + c * GF_PL + row * GF_W);
        ((float4*)rows[3 + c])[t] = py[t];
    }
    __syncthreads();

    const int c0 = t * 4;
    float acc[18];
#pragma unroll
    for (int ch = 0; ch < 18; ++ch) acc[ch] = 0.0f;

    auto tap = [&](int k, float sgn) {
        const float x0 = rows[0][k], x1 = rows[1][k], x2 = rows[2][k];
        const float y0 = rows[3][k], y1 = rows[4][k], y2 = rows[5][k];
        acc[0] += sgn * x0;  acc[1] += sgn * x1;  acc[2] += sgn * x2;
        acc[3] += sgn * y0;  acc[4] += sgn * y1;  acc[5] += sgn * y2;
        acc[6] += sgn * x0 * x0; acc[7] += sgn * x1 * x1; acc[8] += sgn * x2 * x2;
        acc[9]  += sgn * y0 * x0; acc[10] += sgn * y0 * x1; acc[11] += sgn * y0 * x2;
        acc[12] += sgn * y1 * x0; acc[13] += sgn * y1 * x1; acc[14] += sgn * y1 * x2;
        acc[15] += sgn * y2 * x0; acc[16] += sgn * y2 * x1; acc[17] += sgn * y2 * x2;
    };

    int lo = c0 - GF_R; if (lo < 0) lo = 0;
    int hi = c0 + GF_R; if (hi > GF_W - 1) hi = GF_W - 1;
    for (int k = lo; k <= hi; ++k) tap(k, 1.0f);

    float ov[18][4];
#pragma unroll
    for (int s = 0; s < 4; ++s) {
        const int col = c0 + s;
        if (s > 0) {
            const int ka = col + GF_R;
            if (ka <= GF_W - 1) tap(ka, 1.0f);
            const int kr = col - GF_R - 1;
            if (kr >= 0) tap(kr, -1.0f);
        }
        const float ic = gf_inv_cnt(col, GF_W);
#pragma unroll
        for (int ch = 0; ch < 18; ++ch) ov[ch][s] = acc[ch] * ic;
    }
#pragma unroll
    for (int ch = 0; ch < 18; ++ch) {
        float4 o = make_float4(ov[ch][0], ov[ch][1], ov[ch][2], ov[ch][3]);
        *(float4*)(o18 + ch * GF_PL + row * GF_W + c0) = o;
    }
}

// ---------------------------------------------------------------------------
// K2: vertical box mean of the 18 planes (WMMA) + per-pixel 3x3 solve:
//   var_j = mean_xx_j - mx_j^2 ; A_ij = (mean_yx_ij - my_i*mx_j)/(var_j+eps_j)
//   b_i   = my_i - sum_j A_ij * mx_j
// Output planes: ch 0..8 = A (i*3+j), ch 9..11 = b.
// ---------------------------------------------------------------------------
__global__ void __launch_bounds__(32)
gf_k2_vmean_ab(const float* __restrict__ in18, const float* __restrict__ omega,
               float* __restrict__ ab) {
    const int lane = threadIdx.x;
    const int cb = blockIdx.x * 16;
    const int obase = blockIdx.y * 16;

    v8f acc[18];
    gf_vert_box_tile<18>(in18, obase, cb, lane, acc);

    const int nlo = lane & 15;
    const int khalf = lane >> 4;
    const int col = cb + nlo;

    float eps[3];
#pragma unroll
    for (int j = 0; j < 3; ++j) eps[j] = gf_softplus(omega[j]);

#pragma unroll
    for (int v = 0; v < 8; ++v) {
        const int row = obase + v + 8 * khalf;
        const float ic = gf_inv_cnt(row, GF_H);
        const int rowoff = row * GF_W + col;   // lane-varying i32
        float mx[3], my[3], ivr[3];
#pragma unroll
        for (int j = 0; j < 3; ++j) mx[j] = acc[j][v] * ic;
#pragma unroll
        for (int i = 0; i < 3; ++i) my[i] = acc[3 + i][v] * ic;
#pragma unroll
        for (int j = 0; j < 3; ++j)
            ivr[j] = 1.0f / (acc[6 + j][v] * ic - mx[j] * mx[j] + eps[j]);
#pragma unroll
        for (int i = 0; i < 3; ++i) {
            float bi = my[i];
#pragma unroll
            for (int j = 0; j < 3; ++j) {
                const float cov = acc[9 + i * 3 + j][v] * ic - my[i] * mx[j];
                const float Aij = cov * ivr[j];
                bi -= Aij * mx[j];
                float* abp = ab + (i * 3 + j) * GF_PL;   // uniform base
                abp[rowoff] = Aij;
            }
            float* bbp = ab + (9 + i) * GF_PL;           // uniform base
            bbp[rowoff] = bi;
        }
    }
}

// ---------------------------------------------------------------------------
// K3: horizontal box mean of the 12 A/b planes for one row.
// ---------------------------------------------------------------------------
__global__ void __launch_bounds__(256)
gf_k3_hmean12(const float* __restrict__ in12, float* __restrict__ o12) {
    __shared__ float rows[12][GF_W];
    const int row = blockIdx.x;
    const int t = threadIdx.x;
#pragma unroll
    for (int ch = 0; ch < 12; ++ch) {
        const float4* p = (const float4*)(in12 + ch * GF_PL + row * GF_W);
        ((float4*)rows[ch])[t] = p[t];
    }
    __syncthreads();

    const int c0 = t * 4;
    float acc[12];
#pragma unroll
    for (int ch = 0; ch < 12; ++ch) acc[ch] = 0.0f;

    auto tap = [&](int k, float sgn) {
#pragma unroll
        for (int ch = 0; ch < 12; ++ch) acc[ch] += sgn * rows[ch][k];
    };

    int lo = c0 - GF_R; if (lo < 0) lo = 0;
    int hi = c0 + GF_R; if (hi > GF_W - 1) hi = GF_W - 1;
    for (int k = lo; k <= hi; ++k) tap(k, 1.0f);

    float ov[12][4];
#pragma unroll
    for (int s = 0; s < 4; ++s) {
        const int col = c0 + s;
        if (s > 0) {
            const int ka = col + GF_R;
            if (ka <= GF_W - 1) tap(ka, 1.0f);
            const int kr = col - GF_R - 1;
            if (kr >= 0) tap(kr, -1.0f);
        }
        const float ic = gf_inv_cnt(col, GF_W);
#pragma unroll
        for (int ch = 0; ch < 12; ++ch) ov[ch][s] = acc[ch] * ic;
    }
#pragma unroll
    for (int ch = 0; ch < 12; ++ch) {
        float4 o = make_float4(ov[ch][0], ov[ch][1], ov[ch][2], ov[ch][3]);
        *(float4*)(o12 + ch * GF_PL + row * GF_W + c0) = o;
    }
}

// ---------------------------------------------------------------------------
// K4: vertical box mean of A/b (WMMA) + combine: out_i = sum_j mA_ij*x_j + mb_i
// ---------------------------------------------------------------------------
__global__ void __launch_bounds__(32)
gf_k4_vmean_out(const float* __restrict__ ab12, const float* __restrict__ xn,
                float* __restrict__ outn) {
    const int lane = threadIdx.x;
    const int cb = blockIdx.x * 16;
    const int obase = blockIdx.y * 16;

    v8f acc[12];
    gf_vert_box_tile<12>(ab12, obase, cb, lane, acc);

    const int nlo = lane & 15;
    const int khalf = lane >> 4;
    const int col = cb + nlo;

#pragma unroll
    for (int v = 0; v < 8; ++v) {
        const int row = obase + v + 8 * khalf;
        const float ic = gf_inv_cnt(row, GF_H);
        const int rowoff = row * GF_W + col;   // lane-varying i32
        float xv[3];
#pragma unroll
        for (int j = 0; j < 3; ++j) {
            const float* xp = xn + j * GF_PL;  // uniform base
            xv[j] = xp[rowoff];
        }
#pragma unroll
        for (int i = 0; i < 3; ++i) {
            float o = acc[9 + i][v] * ic;      // mean_b
#pragma unroll
            for (int j = 0; j < 3; ++j)
                o += (acc[i * 3 + j][v] * ic) * xv[j];  // mean_A * x
            float* op = outn + i * GF_PL;      // uniform base
            op[rowoff] = o;
        }
    }
}

extern "C" void kernel_launch(void* const* d_in, const int* in_sizes, int n_in,
                              void* d_out, int out_size, void* d_ws, size_t ws_size,
                              hipStream_t stream) {
    (void)in_sizes; (void)n_in; (void)out_size; (void)ws_size;
    const float* y = (const float*)d_in[0];      // (4,3,1024,1024)
    const float* x = (const float*)d_in[1];      // (4,3,1024,1024)
    const float* omega = (const float*)d_in[2];  // (3,)
    float* out = (float*)d_out;                  // (4,3,1024,1024)

    const size_t HW = (size_t)GF_PL;
    // Workspace layout (per-batch reuse): 18 planes + 12 planes = 126 MB.
    float* buf18 = (float*)d_ws;          // K1 out / K2 in; reused as ab2
    float* ab1 = buf18 + 18 * HW;         // K2 out / K3 in
    float* ab2 = buf18;                   // K3 out / K4 in (buf18 dead by then)

    const dim3 tiles(GF_W / 16, GF_H / 16);
    for (int n = 0; n < 4; ++n) {
        const float* xn = x + (size_t)n * 3 * HW;
        const float* yn = y + (size_t)n * 3 * HW;
        float* outn = out + (size_t)n * 3 * HW;
        gf_k1_hmean18<<<GF_H, 256, 0, stream>>>(xn, yn, buf18);
        gf_k2_vmean_ab<<<tiles, 32, 0, stream>>>(buf18, omega, ab1);
        gf_k3_hmean12<<<GF_H, 256, 0, stream>>>(ab1, ab2);
        gf_k4_vmean_out<<<tiles, 32, 0, stream>>>(ab2, xn, outn);
    }
}